// ChannelWiseTensorProduct_1151051235359
// MI455X (gfx1250) — compile-verified
//
#include <hip/hip_runtime.h>

typedef __attribute__((ext_vector_type(2))) float v2f;
typedef __attribute__((ext_vector_type(8))) float v8f;

#define NPATH   24
#define CG_TOTAL 1826     // sum over paths of (2l1+1)(2l2+1)(2l3+1)
#define NM      388       // sum over paths of (2l1+1)(2l3+1)
#define MUL     128
#define DIM1    1152
#define DIM2    16
#define DIM_OUT 13568
#define WNUM    3072

// Path list (l1 outer, l2 mid, l3 inner), matches the reference PATHS order.
__constant__ int PL1[NPATH] = {0,0,0,0, 1,1,1,1,1,1,1,1,1, 2,2,2,2,2,2,2,2,2,2,2};
__constant__ int PL2[NPATH] = {0,1,2,3, 0,1,1,1,2,2,2,3,3, 0,1,1,1,2,2,2,2,3,3,3};
__constant__ int PL3[NPATH] = {0,1,2,3, 1,0,1,2,1,2,3,2,3, 2,1,2,3,0,1,2,3,1,2,3};

// ---------------- CG setup (recomputed each launch; deterministic) ----------------

__device__ __forceinline__ double dfact(int n) {
    const double F[10] = {1.,1.,2.,6.,24.,120.,720.,5040.,40320.,362880.};
    return F[n];
}

__device__ double su2_cg(int j1, int j2, int j3, int m1, int m2, int m3) {
    if (m3 != m1 + m2) return 0.0;
    if (j3 < abs(j1 - j2) || j3 > j1 + j2) return 0.0;
    double pref = sqrt((2.0*j3 + 1.0) * dfact(j3+j1-j2) * dfact(j3-j1+j2) * dfact(j1+j2-j3)
                       / dfact(j1+j2+j3+1)
                       * dfact(j3+m3) * dfact(j3-m3) * dfact(j1-m1) * dfact(j1+m1)
                       * dfact(j2-m2) * dfact(j2+m2));
    double s = 0.0;
    for (int v = 0; v <= j1 + j2 - j3; ++v) {
        int d0 = v, d1 = j1+j2-j3-v, d2 = j1-m1-v, d3 = j2+m2-v, d4 = j3-j2+m1+v, d5 = j3-j1-m2+v;
        if (d0 < 0 || d1 < 0 || d2 < 0 || d3 < 0 || d4 < 0 || d5 < 0) continue;
        double term = 1.0 / (dfact(d0)*dfact(d1)*dfact(d2)*dfact(d3)*dfact(d4)*dfact(d5));
        s += (v & 1) ? -term : term;
    }
    return pref * s;
}

struct cplx { double re, im; };

// Entry (row,col) of the real->complex change-of-basis Q_l, including the (-i)^l phase.
__device__ cplx qent(int l, int row, int col) {
    int m = row - l;
    const double is2 = 0.7071067811865475244;
    double re = 0.0, im = 0.0;
    if (m < 0) {
        if (col == l - m)      re =  is2;   // col == l+|m|
        else if (col == l + m) im = -is2;   // col == l-|m|
    } else if (m == 0) {
        if (col == l) re = 1.0;
    } else {
        double sg = (m & 1) ? -1.0 : 1.0;
        if (col == l + m)      re = sg * is2;
        else if (col == l - m) im = sg * is2;
    }
    // multiply by (-i)^l
    double r2, i2;
    switch (l & 3) {
        case 0: r2 =  re; i2 =  im; break;
        case 1: r2 =  im; i2 = -re; break;
        case 2: r2 = -re; i2 = -im; break;
        default: r2 = -im; i2 =  re; break;
    }
    return {r2, i2};
}

__global__ __launch_bounds__(256) void cg_setup_kernel(float* __restrict__ cgbuf) {
    int t = blockIdx.x * 256 + threadIdx.x;
    if (t >= CG_TOTAL) return;
    int rem = t, p = 0, d1 = 1, d2 = 1, d3 = 1;
    for (p = 0; p < NPATH; ++p) {
        d1 = 2*PL1[p]+1; d2 = 2*PL2[p]+1; d3 = 2*PL3[p]+1;
        int sz = d1 * d2 * d3;
        if (rem < sz) break;
        rem -= sz;
    }
    int l1 = PL1[p], l2 = PL2[p], l3 = PL3[p];
    int ri = rem / (d2 * d3);
    int rr = rem % (d2 * d3);
    int rj = rr / d3;
    int rk = rr % d3;
    // C[ri,rj,rk] = Re( sum_{a,b,c} Q1[a,ri] Q2[b,rj] conj(Q3[c,rk]) cg(a-l1,b-l2,c-l3) )
    // Q column `col` has nonzero rows only at {col, 2l-col}.
    int al[2] = {ri, 2*l1 - ri}; int na = (al[1] == al[0]) ? 1 : 2;
    int bl[2] = {rj, 2*l2 - rj}; int nb = (bl[1] == bl[0]) ? 1 : 2;
    int cl[2] = {rk, 2*l3 - rk}; int nc = (cl[1] == cl[0]) ? 1 : 2;
    double acc = 0.0;
    for (int ia = 0; ia < na; ++ia)
    for (int ib = 0; ib < nb; ++ib)
    for (int ic = 0; ic < nc; ++ic) {
        int a = al[ia], b = bl[ib], c = cl[ic];
        double cgv = su2_cg(l1, l2, l3, a - l1, b - l2, c - l3);
        if (cgv == 0.0) continue;
        cplx q1 = qent(l1, a, ri), q2 = qent(l2, b, rj), q3 = qent(l3, c, rk);
        double pr = q1.re * q2.re - q1.im * q2.im;
        double pi = q1.re * q2.im + q1.im * q2.re;
        acc += (pr * q3.re + pi * q3.im) * cgv;   // Re(q1 q2 conj(q3)) * cg
    }
    cgbuf[t] = (float)acc;
}

// ---------------- Main kernel: one block (8 waves) per batch row ----------------
//
// Padded-column layout for WMMA B / output mapping (8 col-tiles of 16):
//   tile 0      : l1=0 group, 16 cols (exact, no pad)
//   tiles 1..3  : l1=1 group, 39 cols padded to 48
//   tiles 4..7  : l1=2 group, 51 cols padded to 64
// s_B row-major per group with zero-padded K rows (so garbage A lanes * 0 = 0):
//   group0: base 0,   width 16, 4 rows ; group1: base 64,  width 48, 4 rows
//   group2: base 256, width 64, 8 rows (K=5 split into two WMMAs: rows 0-3, row 4)

__global__ __launch_bounds__(256) void cwtp_kernel(
    const float* __restrict__ x1, const float* __restrict__ x2,
    const float* __restrict__ w,  const float* __restrict__ cg,
    float* __restrict__ out)
{
    __shared__ float s_w[WNUM];
    __shared__ float s_x1[1160];     // 1152 + zeroed tail (keeps K-pad reads finite)
    __shared__ float s_x2[DIM2];
    __shared__ float s_B[768];
    __shared__ int   s_cp[128];      // padded col -> path (-1 = pad)
    __shared__ int   s_ck[128];      // padded col -> k within path
    __shared__ int   s_poff[NPATH];  // output float offset of path block
    __shared__ int   s_pd3[NPATH];

    const int tid = threadIdx.x;
    const int b   = blockIdx.x;

    for (int i = tid; i < WNUM; i += 256) s_w[i] = w[i];
    for (int i = tid; i < 1160; i += 256)
        s_x1[i] = (i < DIM1) ? x1[(size_t)b * DIM1 + i] : 0.0f;
    if (tid < DIM2) s_x2[tid] = x2[(size_t)b * DIM2 + tid];
    for (int i = tid; i < 768; i += 256) s_B[i] = 0.0f;
    if (tid < NPATH) {
        int off = 0;
        for (int q = 0; q < tid; ++q) off += MUL * (2*PL3[q] + 1);
        s_poff[tid] = off;
        s_pd3[tid]  = 2*PL3[tid] + 1;
    }
    if (tid < 128) {
        int grp   = (tid < 16) ? 0 : ((tid < 64) ? 1 : 2);
        int local = tid - ((grp == 0) ? 0 : ((grp == 1) ? 16 : 64));
        int p0    = (grp == 0) ? 0 : ((grp == 1) ? 4 : 13);
        int p1    = (grp == 0) ? 4 : ((grp == 1) ? 13 : 24);
        int pf = -1, k = 0, c = local;
        for (int pp = p0; pp < p1; ++pp) {
            int d3 = 2*PL3[pp] + 1;
            if (c < d3) { pf = pp; k = c; break; }
            c -= d3;
        }
        s_cp[tid] = pf; s_ck[tid] = k;
    }
    __syncthreads();

    // Stage 1: M_p[i,k] = sum_j C_p[i,j,k] * x2[b,j]  -> zero-padded B tiles in LDS
    for (int e = tid; e < NM; e += 256) {
        int rem = e, p = 0, d1 = 1, d2 = 1, d3 = 1;
        for (p = 0; p < NPATH; ++p) {
            d1 = 2*PL1[p]+1; d2 = 2*PL2[p]+1; d3 = 2*PL3[p]+1;
            int n = d1 * d3;
            if (rem < n) break;
            rem -= n;
        }
        int i = rem / d3, k = rem % d3;
        int cgo = 0;
        for (int q = 0; q < p; ++q) cgo += (2*PL1[q]+1)*(2*PL2[q]+1)*(2*PL3[q]+1);
        int x2o = PL2[p] * PL2[p];           // 0,1,4,9 for l2=0..3
        float s = 0.0f;
        for (int j = 0; j < d2; ++j)
            s += cg[cgo + (i*d2 + j)*d3 + k] * s_x2[x2o + j];
        int g      = PL1[p];
        int gBase  = (g == 0) ? 0  : ((g == 1) ? 64 : 256);
        int gW     = (g == 0) ? 16 : ((g == 1) ? 48 : 64);
        int pstart = (g == 0) ? 0  : ((g == 1) ? 4  : 13);
        int cb = 0;
        for (int q = pstart; q < p; ++q) cb += 2*PL3[q] + 1;
        s_B[gBase + i*gW + cb + k] = s;
    }
    __syncthreads();

    // Stage 2: per-group GEMM out[u, cols] = x1[u, i] * M[i, cols] via WMMA f32 16x16x4
    const int wave = tid >> 5;
    const int lane = tid & 31;
    const int half = lane >> 4;     // A: M = lane&15; VGPR0 K=2*half, VGPR1 K=2*half+1
    const int l    = lane & 15;

    for (int j = wave; j < 64; j += 8) {        // 8 u-tiles x 8 col-tiles, uniform per wave
        const int utile = j >> 3;
        const int ct    = j & 7;
        const int g     = (ct == 0) ? 0 : ((ct < 4) ? 1 : 2);
        const int gBase = (g == 0) ? 0   : ((g == 1) ? 64  : 256);
        const int gW    = (g == 0) ? 16  : ((g == 1) ? 48  : 64);
        const int cb    = (g == 0) ? 0   : ((g == 1) ? (ct - 1) * 16 : (ct - 4) * 16);
        const int x1b   = (g == 0) ? 0   : ((g == 1) ? 128 : 512);   // 128*l1^2
        const int d1    = 2*g + 1;
        const int urow  = utile * 16 + l;

        v2f a, bf;
        a.x  = s_x1[x1b + urow * d1 + 2*half];        // A(M=urow, K=2*half)
        a.y  = s_x1[x1b + urow * d1 + 2*half + 1];    // A(M=urow, K=2*half+1)
        // Assumed B 4x16 layout (mirrors A's K split / C-D row striping):
        // VGPR0: rows K=0 (lanes 0-15) / K=2 (lanes 16-31); VGPR1: K=1 / K=3; N=lane&15
        bf.x = s_B[gBase + (2*half)     * gW + cb + l];
        bf.y = s_B[gBase + (2*half + 1) * gW + cb + l];

        v8f c = {0.f, 0.f, 0.f, 0.f, 0.f, 0.f, 0.f, 0.f};
        c = __builtin_amdgcn_wmma_f32_16x16x4_f32(false, a, false, bf, (short)0, c, false, false);
        if (g == 2) {   // K = 5: second WMMA covers i = 4 (rows 5..7 of s_B are zero)
            v2f a2, b2;
            a2.x = s_x1[x1b + urow * 5 + 4 + 2*half];
            a2.y = s_x1[x1b + urow * 5 + 5 + 2*half];
            b2.x = s_B[gBase + (4 + 2*half) * gW + cb + l];
            b2.y = s_B[gBase + (5 + 2*half) * gW + cb + l];
            c = __builtin_amdgcn_wmma_f32_16x16x4_f32(false, a2, false, b2, (short)0, c, false, false);
        }

        // D layout: VGPR r -> (M = r + 8*half, N = lane&15). Scale by w[p,u], scatter-store.
        const size_t outRow = (size_t)b * DIM_OUT;
        const int gp = ct * 16 + l;
        const int p  = s_cp[gp];
        if (p >= 0) {
            const int k    = s_ck[gp];
            const int poff = s_poff[p];
            const int d3p  = s_pd3[p];
            #pragma unroll
            for (int r = 0; r < 8; ++r) {
                int u = utile * 16 + r + 8 * half;
                out[outRow + poff + u * d3p + k] = c[r] * s_w[p * MUL + u];
            }
        }
    }
}

extern "C" void kernel_launch(void* const* d_in, const int* in_sizes, int n_in,
                              void* d_out, int out_size, void* d_ws, size_t ws_size,
                              hipStream_t stream) {
    const float* x1 = (const float*)d_in[0];
    const float* x2 = (const float*)d_in[1];
    const float* w  = (const float*)d_in[2];
    float* out = (float*)d_out;
    float* cg  = (float*)d_ws;                 // 1826 floats of scratch

    const int batch = in_sizes[0] / DIM1;      // 8192

    cg_setup_kernel<<<(CG_TOTAL + 255) / 256, 256, 0, stream>>>(cg);
    cwtp_kernel<<<batch, 256, 0, stream>>>(x1, x2, w, cg, out);
}